// GeneratorBilinear_64226940945049
// MI455X (gfx1250) — compile-verified
//
#include <hip/hip_runtime.h>

typedef __attribute__((ext_vector_type(2))) float v2f;
typedef __attribute__((ext_vector_type(8))) float v8f;

// ---------------------------------------------------------------------------
// Compile-time PGA G(3,0,1) tables (blade order matches the reference BLADES)
// Blade index <-> bitmask (bit d set => e_d present)
// ---------------------------------------------------------------------------
constexpr int MASKC[16]  = {0,1,2,4,8,3,5,9,6,10,12,7,11,13,14,15};
constexpr int IDXOFC[16] = {0,1,2,5,3,6,8,11,4,7,9,12,10,13,14,15};

constexpr int popc4(int x){ return ((x>>0)&1)+((x>>1)&1)+((x>>2)&1)+((x>>3)&1); }
// parity of swaps merging sorted blade a with sorted blade b
constexpr int reorderSign(int a, int b){
  int sw = 0;
  for (int i = 0; i < 4; ++i)
    if ((b >> i) & 1) sw += popc4(a >> (i + 1));
  return (sw & 1) ? -1 : 1;
}

struct Term { unsigned char i, j, k; signed char s; };
struct TermTab { Term t[256]; int n; };

constexpr TermTab makeTab(bool wedge){
  TermTab T{}; T.n = 0;
  for (int i = 0; i < 16; ++i)
    for (int j = 0; j < 16; ++j){
      const int a = MASKC[i], b = MASKC[j];
      // geometric product vanishes iff shared e0 (metric 0); wedge iff any overlap
      if (wedge ? ((a & b) != 0) : ((a & b & 1) != 0)) continue;
      T.t[T.n].i = (unsigned char)i;
      T.t[T.n].j = (unsigned char)j;
      T.t[T.n].k = (unsigned char)IDXOFC[a ^ b];
      T.t[T.n].s = (signed char)reorderSign(a, b);
      T.n++;
    }
  return T;
}
constexpr TermTab GPT = makeTab(false);   // 192 terms
constexpr TermTab WDT = makeTab(true);    // 81 terms

// dual(x)[15-i] = DSIGN(i) * x[i]; undual(y)[i] = DSIGN(i) * y[15-i]
constexpr int DSIGNC(int i){ return reorderSign(MASKC[i], 15 ^ MASKC[i]); }

// per-blade runtime constants (wave-uniform indexing -> constant memory)
__device__ __constant__ int GRADE_C[16] = {0,1,1,1,1,2,2,2,2,2,2,3,3,3,3,4};
// e0-containing blade k -> partner blade k\{e0}; -1 otherwise
__device__ __constant__ int EPART_C[16] = {-1,0,-1,-1,-1,2,3,4,-1,-1,-1,8,9,10,-1,14};

// ---------------------------------------------------------------------------
// Fused kernel: one block = one 16-point tile, 128 threads = 4 waves.
// LDS: Xs 12288 f (X[k][i48][p16], later O[k][c48][p16])
//      Hs 24576 f (H[k][c96][p16], later OutStage[p][pitch 772])
// ---------------------------------------------------------------------------
__global__ __launch_bounds__(128)
void ga_fused_kernel(const float* __restrict__ hidden,
                     const float* __restrict__ refv,
                     const float* __restrict__ w_bili,
                     const float* __restrict__ w_next,
                     float* __restrict__ out)
{
  extern __shared__ float smem[];
  float* Xs = smem;           // 12288 floats
  float* Hs = smem + 12288;   // 24576 floats

  const int tid  = threadIdx.x;
  const int wv   = tid >> 5;
  const int lane = tid & 31;
  const int half = lane >> 4;
  const int lm   = lane & 15;
  const int tile = blockIdx.x;

  // 64 tiles per batch (P=1024, 16 points/tile); pseudoscalar of ref
  const float refps = refv[(tile >> 6) * 16 + 15];

  // ---- phase 1: coalesced load + transpose into X[k][i][p] -----------------
  const float4* gx = (const float4*)(hidden + (size_t)tile * 12288);
  for (int v = tid; v < 3072; v += 128) {
    const float4 q = gx[v];
    const int f   = v << 2;          // flat float index in tile: p*768 + i*16 + k
    const int p   = f / 768;
    const int rem = f - p * 768;
    const int i   = rem >> 4;
    const int k0  = rem & 15;        // multiple of 4
    Xs[((k0+0)*48 + i)*16 + p] = q.x;
    Xs[((k0+1)*48 + i)*16 + p] = q.y;
    Xs[((k0+2)*48 + i)*16 + p] = q.z;
    Xs[((k0+3)*48 + i)*16 + p] = q.w;
  }
  __syncthreads();

  // ---- phase 2: equi_linear #1 (96 <- 48) per blade, f32 WMMA --------------
  #pragma unroll 1
  for (int u = wv; u < 96; u += 4) {
    const int mt = u % 6;            // output M-tile (16 channels)
    const int k  = u / 6;            // blade
    const int g  = GRADE_C[k];
    const int o  = mt * 16 + lm;     // A-row this lane owns
    v8f acc = {0.f,0.f,0.f,0.f,0.f,0.f,0.f,0.f};
    #pragma unroll
    for (int ks = 0; ks < 12; ++ks) {
      const int i0 = ks*4 + half*2;  // A: lanes0-15 K=0,1; lanes16-31 K=2,3
      v2f Av, Bv;
      Av.x = w_bili[(o*48 + i0    )*9 + g];
      Av.y = w_bili[(o*48 + i0 + 1)*9 + g];
      Bv.x = Xs[(k*48 + i0    )*16 + lm];
      Bv.y = Xs[(k*48 + i0 + 1)*16 + lm];
      acc = __builtin_amdgcn_wmma_f32_16x16x4_f32(false, Av, false, Bv,
                                                  (short)0, acc, false, false);
    }
    const int kp = EPART_C[k];       // wave-uniform branch (EXEC stays full)
    if (kp >= 0) {
      const int gb = 4 + g;          // e0-map basis slot
      #pragma unroll
      for (int ks = 0; ks < 12; ++ks) {
        const int i0 = ks*4 + half*2;
        v2f Av, Bv;
        Av.x = w_bili[(o*48 + i0    )*9 + gb];
        Av.y = w_bili[(o*48 + i0 + 1)*9 + gb];
        Bv.x = Xs[(kp*48 + i0    )*16 + lm];
        Bv.y = Xs[(kp*48 + i0 + 1)*16 + lm];
        acc = __builtin_amdgcn_wmma_f32_16x16x4_f32(false, Av, false, Bv,
                                                    (short)0, acc, false, false);
      }
    }
    #pragma unroll
    for (int r = 0; r < 8; ++r)      // D layout: M = r + 8*half, N = lm
      Hs[(k*96 + mt*16 + r + half*8)*16 + lm] = acc[r];
  }
  __syncthreads();

  // ---- phase 3: geometric product + equivariant join (VALU) ----------------
  float* Os = Xs;                    // O[k][c48][p16] overwrites X
  #pragma unroll 1
  for (int t = 0; t < 3; ++t) {
    const int pair = tid + t * 128;  // 384 (c,p) pairs
    const int c = pair >> 4;
    const int p = pair & 15;
    float a[16], b[16], r[16];
    // geometric product: lg = ch c, rg = ch 24+c
    #pragma unroll
    for (int i = 0; i < 16; ++i) {
      a[i] = Hs[(i*96 + c     )*16 + p];
      b[i] = Hs[(i*96 + 24 + c)*16 + p];
      r[i] = 0.f;
    }
    #pragma unroll
    for (int q = 0; q < GPT.n; ++q)
      r[GPT.t[q].k] += (float)GPT.t[q].s * a[GPT.t[q].i] * b[GPT.t[q].j];
    #pragma unroll
    for (int i = 0; i < 16; ++i)
      Os[(i*48 + c)*16 + p] = r[i];
    // join: lj = ch 48+c, rj = ch 72+c ; dual / undual are signed reversals
    float da[16], db[16];
    #pragma unroll
    for (int i = 0; i < 16; ++i) {
      da[15-i] = (float)DSIGNC(i) * Hs[(i*96 + 48 + c)*16 + p];
      db[15-i] = (float)DSIGNC(i) * Hs[(i*96 + 72 + c)*16 + p];
      r[i] = 0.f;
    }
    #pragma unroll
    for (int q = 0; q < WDT.n; ++q)
      r[WDT.t[q].k] += (float)WDT.t[q].s * da[WDT.t[q].i] * db[WDT.t[q].j];
    #pragma unroll
    for (int i = 0; i < 16; ++i)
      Os[(i*48 + 24 + c)*16 + p] = (float)DSIGNC(i) * r[15-i] * refps;
  }
  __syncthreads();

  // ---- phase 4: equi_linear #2 (48 <- 48), stage OUT[p][o*16+k], pitch 772 -
  float* St = Hs;                    // 16*772 = 12352 floats, fits in Hs
  #pragma unroll 1
  for (int u = wv; u < 48; u += 4) {
    const int mt = u % 3;
    const int k  = u / 3;
    const int g  = GRADE_C[k];
    const int o  = mt * 16 + lm;
    v8f acc = {0.f,0.f,0.f,0.f,0.f,0.f,0.f,0.f};
    #pragma unroll
    for (int ks = 0; ks < 12; ++ks) {
      const int i0 = ks*4 + half*2;
      v2f Av, Bv;
      Av.x = w_next[(o*48 + i0    )*9 + g];
      Av.y = w_next[(o*48 + i0 + 1)*9 + g];
      Bv.x = Os[(k*48 + i0    )*16 + lm];
      Bv.y = Os[(k*48 + i0 + 1)*16 + lm];
      acc = __builtin_amdgcn_wmma_f32_16x16x4_f32(false, Av, false, Bv,
                                                  (short)0, acc, false, false);
    }
    const int kp = EPART_C[k];
    if (kp >= 0) {
      const int gb = 4 + g;
      #pragma unroll
      for (int ks = 0; ks < 12; ++ks) {
        const int i0 = ks*4 + half*2;
        v2f Av, Bv;
        Av.x = w_next[(o*48 + i0    )*9 + gb];
        Av.y = w_next[(o*48 + i0 + 1)*9 + gb];
        Bv.x = Os[(kp*48 + i0    )*16 + lm];
        Bv.y = Os[(kp*48 + i0 + 1)*16 + lm];
        acc = __builtin_amdgcn_wmma_f32_16x16x4_f32(false, Av, false, Bv,
                                                    (short)0, acc, false, false);
      }
    }
    #pragma unroll
    for (int r = 0; r < 8; ++r)
      St[lm*772 + (mt*16 + r + half*8)*16 + k] = acc[r];
  }
  __syncthreads();

  // ---- phase 5: coalesced store --------------------------------------------
  float4* gout = (float4*)(out + (size_t)tile * 12288);
  for (int v = tid; v < 3072; v += 128) {
    const int p  = v / 192;
    const int c4 = v - p * 192;
    const float* s = &St[p*772 + c4*4];   // 772*4 bytes is 16B-aligned
    float4 q; q.x = s[0]; q.y = s[1]; q.z = s[2]; q.w = s[3];
    gout[v] = q;
  }
}

// ---------------------------------------------------------------------------
extern "C" void kernel_launch(void* const* d_in, const int* in_sizes, int n_in,
                              void* d_out, int out_size, void* d_ws, size_t ws_size,
                              hipStream_t stream) {
  const float* hidden = (const float*)d_in[0];   // [64,1024,48,16] f32
  const float* refv   = (const float*)d_in[1];   // [64,1,1,16]     f32
  const float* w_bili = (const float*)d_in[2];   // [96,48,9]       f32
  const float* w_next = (const float*)d_in[3];   // [48,48,9]       f32
  float* out = (float*)d_out;                    // [64,1024,48,16] f32

  const int npts  = in_sizes[0] / 768;           // 65536 points
  const int tiles = npts / 16;                   // 4096 tiles

  const size_t shmem = (size_t)36864 * sizeof(float);   // 144 KB
  (void)hipFuncSetAttribute((const void*)ga_fused_kernel,
                            hipFuncAttributeMaxDynamicSharedMemorySize,
                            (int)shmem);
  hipLaunchKernelGGL(ga_fused_kernel, dim3(tiles), dim3(128), shmem, stream,
                     hidden, refv, w_bili, w_next, out);
}